// ScatterSum_56805237457287
// MI455X (gfx1250) — compile-verified
//
#include <hip/hip_runtime.h>
#include <hip/hip_bf16.h>
#include <stdint.h>

// Scatter-sum: out[index[e], :] += src[e, :], D_FEAT = 64, f32.
// HBM-bound (~448 MB total traffic -> ~19us at 23.3 TB/s on MI455X).
// Pipeline: gfx1250 async global->LDS copies (double buffered) feed
// hardware f32 atomics (which stay resident in the 192MB L2).

#define TPB 256          // 8 wave32s
#define LANES_PER_EDGE 16 // 16 lanes x float4 = 64 floats
#define EDGES_PER_TILE (TPB / LANES_PER_EDGE) // 16
#define GRID_BLOCKS 12500

__global__ void ScatterSum_zero_kernel(float* __restrict__ out, long long n) {
    long long i = ((long long)blockIdx.x * blockDim.x + threadIdx.x) * 4;
    if (i + 3 < n) {
        float4 z; z.x = 0.f; z.y = 0.f; z.z = 0.f; z.w = 0.f;
        *(float4*)(out + i) = z;
    } else {
        for (; i < n; ++i) out[i] = 0.f;
    }
}

__global__ __launch_bounds__(TPB) void ScatterSum_56805237457287_kernel(
    const float* __restrict__ src,
    const int*   __restrict__ index,
    float*       __restrict__ out,
    long long n_edges,
    long long ntiles) {

    __shared__ float buf[2][TPB * 4]; // 2 x 4KB double buffer

    const int t       = threadIdx.x;
    const int eInTile = t >> 4;   // 0..15 : which edge of the tile
    const int chunk   = t & 15;   // 0..15 : which float4 of the 64 feats
    const long long stride = gridDim.x;
    long long tile = blockIdx.x;

    // 32-bit LDS byte offsets for this lane's slot in each buffer
    // (generic 'shared' pointer low 32 bits == LDS offset on AMDGCN).
    const uint32_t lds0 = (uint32_t)(uintptr_t)(&buf[0][t * 4]);
    const uint32_t lds1 = (uint32_t)(uintptr_t)(&buf[1][t * 4]);

    if (tile < ntiles) {
        // ---- prologue: issue async copy of tile 0, prefetch its node id ----
        {
            uint64_t ga = (uint64_t)(const void*)(src +
                (tile * EDGES_PER_TILE + eInTile) * 64 + chunk * 4);
            asm volatile("global_load_async_to_lds_b128 %0, %1, off"
                         :: "v"(lds0), "v"(ga) : "memory");
        }
        int curNode = index[tile * EDGES_PER_TILE + eInTile];
        int parity = 0;

        while (true) {
            long long next = tile + stride;
            const bool more = (next < ntiles);
            int nextNode = 0;

            if (more) {
                // issue async copy of the NEXT tile into the other buffer
                uint32_t l = parity ? lds0 : lds1;
                uint64_t g = (uint64_t)(const void*)(src +
                    (next * EDGES_PER_TILE + eInTile) * 64 + chunk * 4);
                asm volatile("global_load_async_to_lds_b128 %0, %1, off"
                             :: "v"(l), "v"(g) : "memory");
                nextNode = index[next * EDGES_PER_TILE + eInTile];
                // current tile done, next tile still in flight
                asm volatile("s_wait_asynccnt 0x1" ::: "memory");
            } else {
                asm volatile("s_wait_asynccnt 0x0" ::: "memory");
            }

            // ---- consume current tile from LDS ----
            const float* p = &buf[parity][t * 4];
            float4 v = *(const float4*)p;
            float* o = out + (long long)curNode * 64 + chunk * 4;
            unsafeAtomicAdd(o + 0, v.x);
            unsafeAtomicAdd(o + 1, v.y);
            unsafeAtomicAdd(o + 2, v.z);
            unsafeAtomicAdd(o + 3, v.w);

            if (!more) break;
            tile = next;
            curNode = nextNode;
            parity ^= 1;
        }
    }

    // ---- tail edges (n_edges % 16), direct path; empty for 1.6M edges ----
    const long long tail_start = ntiles * EDGES_PER_TILE;
    for (long long e = tail_start + blockIdx.x; e < n_edges; e += stride) {
        if (t < LANES_PER_EDGE) {
            int node = index[e];
            float4 v = *(const float4*)(src + e * 64 + t * 4);
            float* o = out + (long long)node * 64 + t * 4;
            unsafeAtomicAdd(o + 0, v.x);
            unsafeAtomicAdd(o + 1, v.y);
            unsafeAtomicAdd(o + 2, v.z);
            unsafeAtomicAdd(o + 3, v.w);
        }
    }
}

extern "C" void kernel_launch(void* const* d_in, const int* in_sizes, int n_in,
                              void* d_out, int out_size, void* d_ws, size_t ws_size,
                              hipStream_t stream) {
    const float* src   = (const float*)d_in[0];
    const int*   index = (const int*)d_in[1];
    float*       out   = (float*)d_out;

    const long long n_edges = (long long)in_sizes[1];
    const long long n_out   = (long long)out_size;

    // 1) zero the (poisoned) output
    {
        long long n4 = (n_out + 3) / 4;
        int blocks = (int)((n4 + TPB - 1) / TPB);
        if (blocks < 1) blocks = 1;
        ScatterSum_zero_kernel<<<blocks, TPB, 0, stream>>>(out, n_out);
    }

    // 2) pipelined async-LDS scatter-add
    {
        long long ntiles = n_edges / EDGES_PER_TILE;
        long long grid = ntiles < GRID_BLOCKS ? (ntiles > 0 ? ntiles : 1)
                                              : GRID_BLOCKS;
        ScatterSum_56805237457287_kernel<<<(int)grid, TPB, 0, stream>>>(
            src, index, out, n_edges, ntiles);
    }
}